// ProjCoCoLossTry_5111011083012
// MI455X (gfx1250) — compile-verified
//
#include <hip/hip_runtime.h>
#include <math.h>

// ---------------------------------------------------------------------------
// Reference shape constants
//   4 modalities, each (B=2, C=128, 64^3 pixels). Embedding rows (E) are
//   ordered e = modality*2 + b,  e in [0,8).  Stage-1 output H is stored
//   linearly as [modality][b][c]  (1024 floats) in d_ws.
// ---------------------------------------------------------------------------
#define P_PIX (64 * 64 * 64)   // 262144 pixels
#define C_DIM 128
#define N_ROWS 1024            // 4 modalities * 2 batch * 128 channels

typedef __attribute__((ext_vector_type(2))) float v2f;
typedef __attribute__((ext_vector_type(4))) float v4f;
typedef __attribute__((ext_vector_type(8))) float v8f;

// ---------------------------------------------------------------------------
// Kernel 1: bandwidth-bound streaming reduction.
//   H[r] = dot(x_row(r), w_pix),  r = modality*256 + b*128 + c
// 1.07 GB of once-touched input at 23.3 TB/s -> ~46 us floor.
//  - modality stream: non-temporal b128 loads (TH=NT) so the 1 GB stream does
//    not displace w_pix (1 MB, reused by all 1024 rows) from L2/WGP$.
//  - v4f accumulator = 4 independent FMA chains; unroll-4 lets the compiler
//    clause 4 outstanding b128 loads per wave for latency hiding.
// Deterministic (fixed in-thread order + fixed LDS tree), no atomics.
// ---------------------------------------------------------------------------
__global__ __launch_bounds__(256) void pix_reduce_kernel(
    const float* __restrict__ t1, const float* __restrict__ t1ce,
    const float* __restrict__ t2, const float* __restrict__ flair,
    const float* __restrict__ w_pix, float* __restrict__ H)
{
    const int r = blockIdx.x;                 // 0..1023
    const float* base;
    switch (r >> 8) {                         // 256 rows per modality
        case 0:  base = t1;    break;
        case 1:  base = t1ce;  break;
        case 2:  base = t2;    break;
        default: base = flair; break;
    }
    const v4f* __restrict__ x4 =
        reinterpret_cast<const v4f*>(base + (size_t)(r & 255) * P_PIX);
    const v4f* __restrict__ w4 = reinterpret_cast<const v4f*>(w_pix);

    v4f acc = {0.0f, 0.0f, 0.0f, 0.0f};
    const int n4 = P_PIX / 4;                 // 65536 v4f per row
    #pragma unroll 4
    for (int i = threadIdx.x; i < n4; i += 256) {
        v4f a = __builtin_nontemporal_load(&x4[i]);  // once-through stream: TH=NT
        v4f w = w4[i];                                // hot in L2 across all rows
        acc += a * w;                                 // 4 independent chains
    }
    // fixed combine order -> bitwise deterministic
    float partial = (acc.x + acc.y) + (acc.z + acc.w);

    __shared__ float red[256];
    red[threadIdx.x] = partial;
    __syncthreads();
    for (int s = 128; s > 0; s >>= 1) {
        if ((int)threadIdx.x < s) red[threadIdx.x] += red[threadIdx.x + s];
        __syncthreads();
    }
    if (threadIdx.x == 0) H[r] = red[0];
}

// ---------------------------------------------------------------------------
// Kernel 2: one wave32 does the whole MLP head + loss with fp32 WMMA.
//   A = H padded to 16x128 (rows 8..15 zero)
//   R = relu(A @ w1 + b1)      via v_wmma_f32_16x16x4_f32 (8 Ntiles x 32 Ksteps)
//   E = R @ w2 + b2            via v_wmma_f32_16x16x4_f32
//   loss = sum_i D[i, i^1] / sum_{j: j/2 != i/2} D[i, j],  D = pairwise L1
//
// VGPR layouts (ISA 7.12.2, wave32):
//   A 16x4 f32 : lane L<16 -> {A[L][k0], A[L][k0+1]},  L>=16 -> K+=2
//   B 4x16 f32 : lane L<16 -> {B[k0][L], B[k0+1][L]},  L>=16 -> K+=2, N=L-16
//   C/D 16x16  : v8f, vgpr r = row (lanes>=16: row r+8), N = lane%16
// EXEC is all-ones around every WMMA (no divergence inside the tile loops).
// ---------------------------------------------------------------------------
__global__ __launch_bounds__(32) void mlp_loss_wmma_kernel(
    const float* __restrict__ Hraw,   // [4][2][128] from kernel 1
    const float* __restrict__ w1, const float* __restrict__ b1,
    const float* __restrict__ w2, const float* __restrict__ b2,
    float* __restrict__ out)
{
    __shared__ float As[16][C_DIM];   // layer-1 input, rows 8..15 = 0
    __shared__ float Rs[16][C_DIM];   // layer-1 output (relu'd)
    __shared__ float Es[8][C_DIM];    // final embeddings (rows 0..7 only)
    __shared__ float contrib[8];

    const int lane = threadIdx.x;     // 0..31, single wave

    // Stage H into LDS in E-row order: e = m*2 + b  ->  Hraw[m*256 + b*128 + c]
    for (int idx = lane; idx < 16 * C_DIM; idx += 32) {
        const int row = idx >> 7;
        const int c   = idx & 127;
        float v = 0.0f;
        if (row < 8) v = Hraw[(row >> 1) * 256 + (row & 1) * 128 + c];
        As[row][c] = v;
    }
    __syncthreads();

    const int mrow  = lane & 15;          // A-matrix row for this lane
    const int khalf = (lane >> 4) * 2;    // K sub-offset: 0 (lo half) / 2 (hi)
    const int ntn   = lane & 15;          // N within tile for B / C / D
    const int rbase = (lane >> 4) * 8;    // D rows: 0..7 (lo) or 8..15 (hi)

    // ---------------- layer 1: Rs = relu(As @ w1 + b1) ----------------
    for (int nt = 0; nt < 8; ++nt) {
        v8f acc = {};
        for (int kk = 0; kk < 32; ++kk) {
            const int kb = kk * 4 + khalf;
            const int nn = nt * 16 + ntn;
            v2f a, b;
            a.x = As[mrow][kb];
            a.y = As[mrow][kb + 1];
            b.x = w1[kb * C_DIM + nn];
            b.y = w1[(kb + 1) * C_DIM + nn];
            acc = __builtin_amdgcn_wmma_f32_16x16x4_f32(
                false, a, false, b, (short)0, acc, false, false);
        }
        const int col = nt * 16 + ntn;
        const float bias = b1[col];
        #pragma unroll
        for (int rr = 0; rr < 8; ++rr) {
            float v = acc[rr] + bias;
            Rs[rbase + rr][col] = v > 0.0f ? v : 0.0f;
        }
    }
    __syncthreads();

    // ---------------- layer 2: Es = Rs @ w2 + b2 (rows 0..7) ----------------
    for (int nt = 0; nt < 8; ++nt) {
        v8f acc = {};
        for (int kk = 0; kk < 32; ++kk) {
            const int kb = kk * 4 + khalf;
            const int nn = nt * 16 + ntn;
            v2f a, b;
            a.x = Rs[mrow][kb];
            a.y = Rs[mrow][kb + 1];
            b.x = w2[kb * C_DIM + nn];
            b.y = w2[(kb + 1) * C_DIM + nn];
            acc = __builtin_amdgcn_wmma_f32_16x16x4_f32(
                false, a, false, b, (short)0, acc, false, false);
        }
        const int col = nt * 16 + ntn;
        if (lane < 16) {                  // keep only real rows 0..7
            const float bias = b2[col];
            #pragma unroll
            for (int rr = 0; rr < 8; ++rr) Es[rr][col] = acc[rr] + bias;
        }
    }
    __syncthreads();

    // ---------------- contrastive L1 loss ----------------
    if (lane < 8) {
        const int i = lane;
        float numer = 0.0f, denom = 0.0f;
        for (int j = 0; j < 8; ++j) {
            if (j == i) continue;
            float d = 0.0f;
            for (int c = 0; c < C_DIM; ++c) d += fabsf(Es[i][c] - Es[j][c]);
            if (j == (i ^ 1))              numer  = d;   // positive (same modality)
            else if ((j >> 1) != (i >> 1)) denom += d;   // 6 negatives
        }
        contrib[i] = numer / denom;
    }
    __syncthreads();
    if (lane == 0) {
        float s = 0.0f;
        for (int i = 0; i < 8; ++i) s += contrib[i];
        out[0] = s;
    }
}

// ---------------------------------------------------------------------------
// Host-side launch (graph-capture safe: kernels on `stream` only, scratch in
// d_ws, every output/scratch location fully overwritten each call).
// ---------------------------------------------------------------------------
extern "C" void kernel_launch(void* const* d_in, const int* in_sizes, int n_in,
                              void* d_out, int out_size, void* d_ws, size_t ws_size,
                              hipStream_t stream) {
    const float* t1    = (const float*)d_in[0];
    const float* t1ce  = (const float*)d_in[1];
    const float* t2    = (const float*)d_in[2];
    const float* flair = (const float*)d_in[3];
    const float* w_pix = (const float*)d_in[4];
    const float* w1    = (const float*)d_in[5];
    const float* b1    = (const float*)d_in[6];
    const float* w2    = (const float*)d_in[7];
    const float* b2    = (const float*)d_in[8];

    float* H = (float*)d_ws;   // 1024 floats of scratch

    pix_reduce_kernel<<<N_ROWS, 256, 0, stream>>>(t1, t1ce, t2, flair, w_pix, H);
    mlp_loss_wmma_kernel<<<1, 32, 0, stream>>>(H, w1, b1, w2, b2, (float*)d_out);
}